// ISET_Layer_1451698946638
// MI455X (gfx1250) — compile-verified
//
#include <hip/hip_runtime.h>
#include <hip/hip_bf16.h>

typedef __attribute__((ext_vector_type(2))) float v2f;
typedef __attribute__((ext_vector_type(8))) float v8f;
typedef unsigned int u32x4 __attribute__((ext_vector_type(4)));
typedef int i32x4 __attribute__((ext_vector_type(4)));
typedef int i32x8 __attribute__((ext_vector_type(8)));

#define V8ZERO {0.f,0.f,0.f,0.f,0.f,0.f,0.f,0.f}

#if __has_include(<hip/amd_detail/amd_gfx1250_TDM.h>)
#define TDM_SIX_ARGS 1
#else
#define TDM_SIX_ARGS 0
#endif

__device__ __forceinline__ float leaky(float x) { return x > 0.0f ? x : 0.01f * x; }

__device__ __forceinline__ v8f wmma4(v2f a, v2f b, v8f c) {
  // D = A(16x4 f32) * B(4x16 f32) + C(16x16 f32)
  return __builtin_amdgcn_wmma_f32_16x16x4_f32(false, a, false, b, (short)0, c, false, false);
}

// ---------------------------------------------------------------------------
// Tensor Data Mover: 2D f32 tile (tile_d0 contiguous elems x tile_d1 rows)
// from a row-major tensor (row stride stride0 elems) into LDS at lds_byte_off.
__device__ __forceinline__ void tdm_load_2d_f32(
    const void* gaddr, unsigned lds_byte_off,
    unsigned tensor_d0, unsigned tensor_d1,
    unsigned tile_d0, unsigned tile_d1, unsigned long long stride0_elems)
{
  unsigned long long ga = (unsigned long long)(size_t)gaddr;
  u32x4 g0;
  g0.x = 1u;                                                  // count=1, user D#
  g0.y = lds_byte_off;                                        // LDS dest (bytes)
  g0.z = (unsigned)ga;                                        // global addr lo32
  g0.w = (unsigned)((ga >> 32) & 0x1FFFFFFull) | (2u << 30);  // addr[56:32] | type=2
  i32x8 g1;
  g1[0] = (int)(2u << 16);                                    // data_size = 4B
  g1[1] = (int)((tensor_d0 & 0xFFFFu) << 16);                 // tensor_dim0 lo16
  g1[2] = (int)((tensor_d0 >> 16) | ((tensor_d1 & 0xFFFFu) << 16));
  g1[3] = (int)((tensor_d1 >> 16) | (tile_d0 << 16));         // tensor_dim1 hi | tile_dim0
  g1[4] = (int)(tile_d1 & 0xFFFFu);                           // tile_dim1 (tile_dim2 = 0)
  g1[5] = (int)(stride0_elems & 0xFFFFFFFFull);               // dim0 stride lo32
  g1[6] = (int)((stride0_elems >> 32) & 0xFFFFull);           // dim0 stride hi16
  g1[7] = 0;
  i32x4 zz = {0, 0, 0, 0};
#if TDM_SIX_ARGS
  i32x8 z8 = {0, 0, 0, 0, 0, 0, 0, 0};
  __builtin_amdgcn_tensor_load_to_lds(g0, g1, zz, zz, z8, 0);
#else
  __builtin_amdgcn_tensor_load_to_lds(g0, g1, zz, zz, 0);
#endif
}

static constexpr int NL = 4096, NR = 8192, EL = 65536, ER = 262144;

// ---------------------------------------------------------------------------
__global__ void k_zero(float* __restrict__ p, int n) {
  int i = blockIdx.x * 256 + threadIdx.x;
  if (i < n) p[i] = 0.0f;
}

// ---------------------------------------------------------------------------
// q = leaky(h@Wq), k = leaky(h@Wk), v = h@Wv   (h: [N,64], W: [64,64])
__global__ __launch_bounds__(128) void k_qkv(
    const float* __restrict__ h,
    const float* __restrict__ Wq, const float* __restrict__ Wk, const float* __restrict__ Wv,
    float* __restrict__ q, float* __restrict__ k, float* __restrict__ v)
{
  __shared__ float hs[16][64];
  const int tid = threadIdx.x, wave = tid >> 5, lane = tid & 31;
  const int m = lane & 15, kh = (lane >> 4) << 1, chalf = (lane >> 4) << 3;
  const int rowbase = blockIdx.x * 16;

  for (int i = tid; i < 256; i += 128)
    ((float4*)hs)[i] = ((const float4*)(h + (size_t)rowbase * 64))[i];
  __syncthreads();

  const int colbase = wave * 16;
  v8f aq = V8ZERO, ak = V8ZERO, av = V8ZERO;
  for (int k0 = 0; k0 < 64; k0 += 4) {
    v2f a; a.x = hs[m][k0 + kh]; a.y = hs[m][k0 + kh + 1];
    int bi = (k0 + kh) * 64 + colbase + m;
    v2f bq, bk, bv;
    bq.x = Wq[bi]; bq.y = Wq[bi + 64];
    bk.x = Wk[bi]; bk.y = Wk[bi + 64];
    bv.x = Wv[bi]; bv.y = Wv[bi + 64];
    aq = wmma4(a, bq, aq);
    ak = wmma4(a, bk, ak);
    av = wmma4(a, bv, av);
  }
#pragma unroll
  for (int u = 0; u < 8; ++u) {
    int row = rowbase + u + chalf, col = colbase + m;
    q[(size_t)row * 64 + col] = leaky(aq[u]);
    k[(size_t)row * 64 + col] = leaky(ak[u]);
    v[(size_t)row * 64 + col] = av[u];
  }
}

// ---------------------------------------------------------------------------
// Edge block: msg = leaky(z@W1+b1)@W2+b2, w = leaky(msg@Wc1+bc1)@Wc2+bc2,
// atomics into msgsum[dst], xsum[dst] (xrel*w), cnt[dst].
// block: 64 threads (2 waves), each wave owns one 16-edge tile.
template<int EI, int FE>
__global__ __launch_bounds__(64) void k_edge(
    const float* __restrict__ h, const float* __restrict__ x,
    const int* __restrict__ src, const int* __restrict__ dst,
    const float* __restrict__ ef,
    const float* __restrict__ W1, const float* __restrict__ b1,
    const float* __restrict__ W2, const float* __restrict__ b2,
    const float* __restrict__ Wc1, const float* __restrict__ bc1,
    const float* __restrict__ Wc2, const float* __restrict__ bc2,
    float* __restrict__ msgsum, float* __restrict__ xsum, float* __restrict__ cnt)
{
  constexpr int EIP = (EI + 3) & ~3;     // pad K to multiple of 4 (EI % 4 == 2)
  constexpr int KFULL = EI & ~3;
  constexpr int TAILC = EIP - 128 - FE;  // rbf (15) + zero pad
  __shared__ float zb[2][16][EIP];
  __shared__ float hidb[2][16][64];
  __shared__ float msgb[2][16][64];
  __shared__ float xrelb[2][16][3];
  __shared__ float d2b[2][16];
  __shared__ int   seb[2][16], deb[2][16];

  const int tid = threadIdx.x, wave = tid >> 5, lane = tid & 31;
  const int m = lane & 15, kh = (lane >> 4) << 1, chalf = (lane >> 4) << 3;
  const int ebase = blockIdx.x * 32 + wave * 16;

  if (lane < 16) {
    int e = ebase + lane;
    int s = src[e], d = dst[e];
    seb[wave][lane] = s; deb[wave][lane] = d;
    float rx = x[s * 3 + 0] - x[d * 3 + 0];
    float ry = x[s * 3 + 1] - x[d * 3 + 1];
    float rz = x[s * 3 + 2] - x[d * 3 + 2];
    xrelb[wave][lane][0] = rx; xrelb[wave][lane][1] = ry; xrelb[wave][lane][2] = rz;
    d2b[wave][lane] = rx * rx + ry * ry + rz * rz;
  }
  __syncthreads();

  // z = [h[src](64) | h[dst](64) | ef(FE) | rbf(15) | pad] -- wide gathers
  for (int i = lane; i < 16 * 16; i += 32) {
    int er = i >> 4, c = (i & 15) << 2;
    *(float4*)&zb[wave][er][c]      = *(const float4*)&h[(size_t)seb[wave][er] * 64 + c];
    *(float4*)&zb[wave][er][64 + c] = *(const float4*)&h[(size_t)deb[wave][er] * 64 + c];
  }
  for (int i = lane; i < 16 * FE; i += 32)   // ef tile is fully contiguous
    zb[wave][i / FE][128 + (i % FE)] = ef[(size_t)ebase * FE + i];
  for (int i = lane; i < 16 * TAILC; i += 32) {
    int er = i / TAILC, j = i % TAILC;
    float val = 0.0f;
    if (j < 15) {
      float sig = 1.0f;
      for (int t = 0; t < j; ++t) sig *= 1.5f;
      val = expf(-d2b[wave][er] / sig);
    }
    zb[wave][er][128 + FE + j] = val;
  }
  __syncthreads();

  // GEMM1: hid = leaky(z@W1 + b1)   K = EI (guarded tail)
  for (int ct = 0; ct < 4; ++ct) {
    v8f acc = V8ZERO;
    for (int k0 = 0; k0 < KFULL; k0 += 4) {
      v2f a; a.x = zb[wave][m][k0 + kh]; a.y = zb[wave][m][k0 + kh + 1];
      int bi = (k0 + kh) * 64 + ct * 16 + m;
      v2f b; b.x = W1[bi]; b.y = W1[bi + 64];
      acc = wmma4(a, b, acc);
    }
    { // tail: only rows KFULL, KFULL+1 exist; padded z cols are zero
      v2f a; a.x = zb[wave][m][KFULL + kh]; a.y = zb[wave][m][KFULL + kh + 1];
      v2f b;
      if (kh == 0) { int bi = KFULL * 64 + ct * 16 + m; b.x = W1[bi]; b.y = W1[bi + 64]; }
      else         { b.x = 0.0f; b.y = 0.0f; }
      acc = wmma4(a, b, acc);
    }
#pragma unroll
    for (int u = 0; u < 8; ++u) {
      int r = u + chalf, c = ct * 16 + m;
      hidb[wave][r][c] = leaky(acc[u] + b1[c]);
    }
  }

  // GEMM2: msg = hid@W2 + b2   (same-wave LDS produce/consume is in-order)
  for (int ct = 0; ct < 4; ++ct) {
    v8f acc = V8ZERO;
    for (int k0 = 0; k0 < 64; k0 += 4) {
      v2f a; a.x = hidb[wave][m][k0 + kh]; a.y = hidb[wave][m][k0 + kh + 1];
      int bi = (k0 + kh) * 64 + ct * 16 + m;
      v2f b; b.x = W2[bi]; b.y = W2[bi + 64];
      acc = wmma4(a, b, acc);
    }
#pragma unroll
    for (int u = 0; u < 8; ++u) {
      int r = u + chalf, c = ct * 16 + m;
      msgb[wave][r][c] = acc[u] + b2[c];
    }
  }

  // GEMM3: hc = leaky(msg@Wc1 + bc1)  (reuse hidb)
  for (int ct = 0; ct < 4; ++ct) {
    v8f acc = V8ZERO;
    for (int k0 = 0; k0 < 64; k0 += 4) {
      v2f a; a.x = msgb[wave][m][k0 + kh]; a.y = msgb[wave][m][k0 + kh + 1];
      int bi = (k0 + kh) * 64 + ct * 16 + m;
      v2f b; b.x = Wc1[bi]; b.y = Wc1[bi + 64];
      acc = wmma4(a, b, acc);
    }
#pragma unroll
    for (int u = 0; u < 8; ++u) {
      int r = u + chalf, c = ct * 16 + m;
      hidb[wave][r][c] = leaky(acc[u] + bc1[c]);
    }
  }

  // scalar head + coordinate/count atomics
  if (lane < 16) {
    float s = bc2[0];
    for (int c = 0; c < 64; ++c) s += hidb[wave][lane][c] * Wc2[c];
    int d = deb[wave][lane];
    atomicAdd(&xsum[d * 3 + 0], xrelb[wave][lane][0] * s);
    atomicAdd(&xsum[d * 3 + 1], xrelb[wave][lane][1] * s);
    atomicAdd(&xsum[d * 3 + 2], xrelb[wave][lane][2] * s);
    atomicAdd(&cnt[d], 1.0f);
  }
  for (int i = lane; i < 16 * 64; i += 32) {
    int er = i >> 6, c = i & 63;
    atomicAdd(&msgsum[(size_t)deb[wave][er] * 64 + c], msgb[wave][er][c]);
  }
}

// ---------------------------------------------------------------------------
// Flash-style masked cross attention: out = softmax(mask*QK^T - 1000*(1-mask)) @ V
// 128 threads, 16 query rows; 64-key chunks; mask tiles streamed by the TDM
// (double-buffered in LDS, overlapped with the QK WMMA loop).
__global__ __launch_bounds__(128) void k_attn(
    const float* __restrict__ q, const float* __restrict__ kmat, const float* __restrict__ vmat,
    const float* __restrict__ mask, int mld, int mrows, int mtrans,
    float* __restrict__ out, int NK)
{
  __shared__ float qs[16][64];
  __shared__ float S[16][64];
  __shared__ float Mt[2][1024];          // mask tiles: [qr][kl] or [kl][qr]
  __shared__ float rowm[16], rowl[16], rowscale[16];

  const int tid = threadIdx.x, wave = tid >> 5, lane = tid & 31;
  const int m = lane & 15, kh = (lane >> 4) << 1, chalf = (lane >> 4) << 3;
  const int qbase = blockIdx.x * 16;
  const int r8 = tid >> 3, t8 = tid & 7;

  for (int i = tid; i < 256; i += 128)
    ((float4*)qs)[i] = ((const float4*)(q + (size_t)qbase * 64))[i];
  if (tid < 16) { rowm[tid] = -3.0e38f; rowl[tid] = 0.0f; }

  // prologue: TDM-load mask tile for chunk 0 into buffer 0
  if (wave == 0) {
    const float* g = mtrans ? (mask + (size_t)0 * mld + qbase)
                            : (mask + (size_t)qbase * mld + 0);
    tdm_load_2d_f32(g, (unsigned)(size_t)&Mt[0][0], mld, mrows,
                    mtrans ? 16u : 64u, mtrans ? 64u : 16u, (unsigned long long)mld);
  }
  __syncthreads();

  v8f acc = V8ZERO;

  for (int kb = 0; kb < NK; kb += 64) {
    const int cur = (kb >> 6) & 1;
    const bool more = (kb + 64 < NK);
    // issue next chunk's mask tile (overlaps with QK WMMAs below)
    if (more && wave == 0) {
      const float* g = mtrans ? (mask + (size_t)(kb + 64) * mld + qbase)
                              : (mask + (size_t)qbase * mld + (kb + 64));
      tdm_load_2d_f32(g, (unsigned)(size_t)&Mt[cur ^ 1][0], mld, mrows,
                      mtrans ? 16u : 64u, mtrans ? 64u : 16u, (unsigned long long)mld);
    }

    // stage 1: S[16 q][64 keys] raw scores (wave w -> key subtile w)
    v8f s = V8ZERO;
    {
      const int key = kb + wave * 16 + m;
      for (int k0 = 0; k0 < 64; k0 += 4) {
        v2f a; a.x = qs[m][k0 + kh]; a.y = qs[m][k0 + kh + 1];
        v2f b; b.x = kmat[(size_t)key * 64 + k0 + kh]; b.y = kmat[(size_t)key * 64 + k0 + kh + 1];
        s = wmma4(a, b, s);
      }
    }
    // current mask tile must be resident (tensor ops complete in order per wave)
    if (more) __builtin_amdgcn_s_wait_tensorcnt(1);
    else      __builtin_amdgcn_s_wait_tensorcnt(0);
    __syncthreads();
#pragma unroll
    for (int u = 0; u < 8; ++u) {
      int qr = u + chalf, kl = wave * 16 + m;
      float mk = mtrans ? Mt[cur][kl * 16 + qr] : Mt[cur][qr * 64 + kl];
      S[qr][kl] = mk * s[u] - 1000.0f * (1.0f - mk);
    }
    __syncthreads();
    // stage 2: online softmax stats (16 rows x 8 lanes)
    {
      float mx = -3.0e38f;
      for (int j = t8; j < 64; j += 8) mx = fmaxf(mx, S[r8][j]);
      mx = fmaxf(mx, __shfl_xor(mx, 4, 8));
      mx = fmaxf(mx, __shfl_xor(mx, 2, 8));
      mx = fmaxf(mx, __shfl_xor(mx, 1, 8));
      float om = rowm[r8];
      float nm = fmaxf(om, mx);
      float sc = expf(om - nm);
      float ps = 0.0f;
      for (int j = t8; j < 64; j += 8) {
        float p = expf(S[r8][j] - nm);
        S[r8][j] = p;
        ps += p;
      }
      ps += __shfl_xor(ps, 4, 8);
      ps += __shfl_xor(ps, 2, 8);
      ps += __shfl_xor(ps, 1, 8);
      if (t8 == 0) { rowl[r8] = rowl[r8] * sc + ps; rowm[r8] = nm; rowscale[r8] = sc; }
    }
    __syncthreads();
    // stage 3: acc = acc*scale + P @ V   (wave w -> feature cols 16w..16w+15)
    {
#pragma unroll
      for (int u = 0; u < 8; ++u) acc[u] = acc[u] * rowscale[u + chalf];
      const int fcol = wave * 16 + m;
      for (int k0 = 0; k0 < 64; k0 += 4) {
        v2f a; a.x = S[m][k0 + kh]; a.y = S[m][k0 + kh + 1];
        v2f b;
        b.x = vmat[(size_t)(kb + k0 + kh) * 64 + fcol];
        b.y = vmat[(size_t)(kb + k0 + kh + 1) * 64 + fcol];
        acc = wmma4(a, b, acc);
      }
      if (more) __builtin_prefetch(&kmat[(size_t)(kb + 64) * 64], 0, 0);
    }
    __syncthreads();   // S and Mt[cur] reused next chunks
  }

#pragma unroll
  for (int u = 0; u < 8; ++u) {
    int qr = u + chalf;
    out[(size_t)(qbase + qr) * 64 + wave * 16 + m] = acc[u] / fmaxf(rowl[qr], 1e-30f);
  }
}

// ---------------------------------------------------------------------------
// Node update: z=[orig_h|h|aggr|cross], hn = 0.5*(leaky(z@Wn1+bn1)@Wn2+bn2)+0.5*h
__global__ __launch_bounds__(128) void k_node(
    const float* __restrict__ h, const float* __restrict__ x,
    const float* __restrict__ oh, const float* __restrict__ ox,
    const float* __restrict__ cross,
    const float* __restrict__ msgsum, const float* __restrict__ xsum, const float* __restrict__ cnt,
    const float* __restrict__ Wn1, const float* __restrict__ bn1,
    const float* __restrict__ Wn2, const float* __restrict__ bn2,
    float* __restrict__ xout, float* __restrict__ hout)
{
  __shared__ float zb[16][256];
  __shared__ float hidb[16][64];
  const int tid = threadIdx.x, wave = tid >> 5, lane = tid & 31;
  const int m = lane & 15, kh = (lane >> 4) << 1, chalf = (lane >> 4) << 3;
  const int nbase = blockIdx.x * 16;

  for (int i = tid; i < 16 * 64; i += 128) {   // 16 nodes x 64 float4
    int n = i >> 6, c = (i & 63) << 2;
    int gn = nbase + n;
    float4 val;
    if (c < 64)       val = *(const float4*)&oh[(size_t)gn * 64 + c];
    else if (c < 128) val = *(const float4*)&h[(size_t)gn * 64 + (c - 64)];
    else if (c < 192) {
      float inv = 1.0f / fmaxf(cnt[gn], 1.0f);
      val = *(const float4*)&msgsum[(size_t)gn * 64 + (c - 128)];
      val.x *= inv; val.y *= inv; val.z *= inv; val.w *= inv;
    } else            val = *(const float4*)&cross[(size_t)gn * 64 + (c - 192)];
    *(float4*)&zb[n][c] = val;
  }
  if (tid < 16) {
    int gn = nbase + tid;
    float cc = fmaxf(cnt[gn], 1.0f);
    for (int d3 = 0; d3 < 3; ++d3)
      xout[gn * 3 + d3] = 0.25f * ox[gn * 3 + d3] + 0.75f * x[gn * 3 + d3] + xsum[gn * 3 + d3] / cc;
  }
  __syncthreads();

  { // hid = leaky(z@Wn1 + bn1), K=256
    const int ct = wave;
    v8f acc = V8ZERO;
    for (int k0 = 0; k0 < 256; k0 += 4) {
      v2f a; a.x = zb[m][k0 + kh]; a.y = zb[m][k0 + kh + 1];
      int bi = (k0 + kh) * 64 + ct * 16 + m;
      v2f b; b.x = Wn1[bi]; b.y = Wn1[bi + 64];
      acc = wmma4(a, b, acc);
    }
#pragma unroll
    for (int u = 0; u < 8; ++u) {
      int r = u + chalf, c = ct * 16 + m;
      hidb[r][c] = leaky(acc[u] + bn1[c]);
    }
  }
  __syncthreads();
  { // hn = 0.5*(hid@Wn2 + bn2) + 0.5*h
    const int ct = wave;
    v8f acc = V8ZERO;
    for (int k0 = 0; k0 < 64; k0 += 4) {
      v2f a; a.x = hidb[m][k0 + kh]; a.y = hidb[m][k0 + kh + 1];
      int bi = (k0 + kh) * 64 + ct * 16 + m;
      v2f b; b.x = Wn2[bi]; b.y = Wn2[bi + 64];
      acc = wmma4(a, b, acc);
    }
#pragma unroll
    for (int u = 0; u < 8; ++u) {
      int gr = nbase + u + chalf, c = ct * 16 + m;
      hout[(size_t)gr * 64 + c] = 0.5f * (acc[u] + bn2[c]) + 0.5f * h[(size_t)gr * 64 + c];
    }
  }
}

// ---------------------------------------------------------------------------
extern "C" void kernel_launch(void* const* d_in, const int* in_sizes, int n_in,
                              void* d_out, int out_size, void* d_ws, size_t ws_size,
                              hipStream_t stream) {
  (void)in_sizes; (void)n_in; (void)out_size; (void)ws_size;

  const float* h_lig  = (const float*)d_in[0];
  const float* x_lig  = (const float*)d_in[1];
  const float* oh_lig = (const float*)d_in[2];
  const float* ox_lig = (const float*)d_in[3];
  const float* h_rec  = (const float*)d_in[4];
  const float* x_rec  = (const float*)d_in[5];
  const float* oh_rec = (const float*)d_in[6];
  const float* ox_rec = (const float*)d_in[7];
  const float* ef_lig = (const float*)d_in[8];
  const float* ef_rec = (const float*)d_in[9];
  const float* mask   = (const float*)d_in[10];
  const int* lig_src  = (const int*)d_in[11];
  const int* lig_dst  = (const int*)d_in[12];
  const int* rec_src  = (const int*)d_in[13];
  const int* rec_dst  = (const int*)d_in[14];

  const float* We_lig1 = (const float*)d_in[15];
  const float* be_lig1 = (const float*)d_in[16];
  const float* We_lig2 = (const float*)d_in[17];
  const float* be_lig2 = (const float*)d_in[18];
  const float* We_rec1 = (const float*)d_in[19];
  const float* be_rec1 = (const float*)d_in[20];
  const float* We_rec2 = (const float*)d_in[21];
  const float* be_rec2 = (const float*)d_in[22];
  const float* Wq_lig  = (const float*)d_in[23];
  const float* Wk_lig  = (const float*)d_in[24];
  const float* Wv_lig  = (const float*)d_in[25];
  const float* Wq_rec  = (const float*)d_in[26];
  const float* Wk_rec  = (const float*)d_in[27];
  const float* Wv_rec  = (const float*)d_in[28];
  const float* Wn_lig1 = (const float*)d_in[29];
  const float* bn_lig1 = (const float*)d_in[30];
  const float* Wn_lig2 = (const float*)d_in[31];
  const float* bn_lig2 = (const float*)d_in[32];
  const float* Wn_rec1 = (const float*)d_in[33];
  const float* bn_rec1 = (const float*)d_in[34];
  const float* Wn_rec2 = (const float*)d_in[35];
  const float* bn_rec2 = (const float*)d_in[36];
  const float* Wc_lig1 = (const float*)d_in[37];
  const float* bc_lig1 = (const float*)d_in[38];
  const float* Wc_lig2 = (const float*)d_in[39];
  const float* bc_lig2 = (const float*)d_in[40];
  const float* Wc_rec1 = (const float*)d_in[41];
  const float* bc_rec1 = (const float*)d_in[42];
  const float* Wc_rec2 = (const float*)d_in[43];
  const float* bc_rec2 = (const float*)d_in[44];

  float* ws = (float*)d_ws;
  float* q_l      = ws;
  float* k_l      = q_l + (size_t)NL * 64;
  float* v_l      = k_l + (size_t)NL * 64;
  float* q_r      = v_l + (size_t)NL * 64;
  float* k_r      = q_r + (size_t)NR * 64;
  float* v_r      = k_r + (size_t)NR * 64;
  float* cross_l  = v_r + (size_t)NR * 64;
  float* cross_r  = cross_l + (size_t)NL * 64;
  float* msgsum_l = cross_r + (size_t)NR * 64;   // -- zeroed region starts here
  float* msgsum_r = msgsum_l + (size_t)NL * 64;
  float* xsum_l   = msgsum_r + (size_t)NR * 64;
  float* xsum_r   = xsum_l + (size_t)NL * 3;
  float* cnt_l    = xsum_r + (size_t)NR * 3;
  float* cnt_r    = cnt_l + NL;
  const int zero_n = NL * 64 + NR * 64 + NL * 3 + NR * 3 + NL + NR;

  k_zero<<<(zero_n + 255) / 256, 256, 0, stream>>>(msgsum_l, zero_n);

  k_qkv<<<NL / 16, 128, 0, stream>>>(h_lig, Wq_lig, Wk_lig, Wv_lig, q_l, k_l, v_l);
  k_qkv<<<NR / 16, 128, 0, stream>>>(h_rec, Wq_rec, Wk_rec, Wv_rec, q_r, k_r, v_r);

  k_edge<158, 15><<<EL / 32, 64, 0, stream>>>(h_lig, x_lig, lig_src, lig_dst, ef_lig,
      We_lig1, be_lig1, We_lig2, be_lig2, Wc_lig1, bc_lig1, Wc_lig2, bc_lig2,
      msgsum_l, xsum_l, cnt_l);
  k_edge<170, 27><<<ER / 32, 64, 0, stream>>>(h_rec, x_rec, rec_src, rec_dst, ef_rec,
      We_rec1, be_rec1, We_rec2, be_rec2, Wc_rec1, bc_rec1, Wc_rec2, bc_rec2,
      msgsum_r, xsum_r, cnt_r);

  k_attn<<<NL / 16, 128, 0, stream>>>(q_l, k_r, v_r, mask, NR, NL, 0, cross_l, NR);
  k_attn<<<NR / 16, 128, 0, stream>>>(q_r, k_l, v_l, mask, NR, NL, 1, cross_r, NL);

  float* out    = (float*)d_out;
  float* xl_out = out;
  float* hl_out = xl_out + (size_t)NL * 3;
  float* xr_out = hl_out + (size_t)NL * 64;
  float* hr_out = xr_out + (size_t)NR * 3;

  k_node<<<NL / 16, 128, 0, stream>>>(h_lig, x_lig, oh_lig, ox_lig, cross_l,
      msgsum_l, xsum_l, cnt_l, Wn_lig1, bn_lig1, Wn_lig2, bn_lig2, xl_out, hl_out);
  k_node<<<NR / 16, 128, 0, stream>>>(h_rec, x_rec, oh_rec, ox_rec, cross_r,
      msgsum_r, xsum_r, cnt_r, Wn_rec1, bn_rec1, Wn_rec2, bn_rec2, xr_out, hr_out);
}